// SpectralMixtureKernel_3547642986768
// MI455X (gfx1250) — compile-verified
//
#include <hip/hip_runtime.h>
#include <hip/hip_bf16.h>

typedef __attribute__((ext_vector_type(2))) float v2f;
typedef __attribute__((ext_vector_type(8))) float v8f;

#define NN 4096
#define MM 4096

// Constants (double-evaluated, truncated to float)
__device__ __constant__ float kLOG2E        = 1.4426950408889634f;   // log2(e)
__device__ __constant__ float kNEG_2PI2_L2E = -28.477967368262680f;  // -(2*pi^2)*log2(e)
__device__ __constant__ float kPOS_4PI2_L2E = 56.955934736525360f;   // +(4*pi^2)*log2(e)

__device__ __forceinline__ float fexp2(float x) {
#if defined(__HIP_DEVICE_COMPILE__) && __has_builtin(__builtin_amdgcn_exp2f)
  return __builtin_amdgcn_exp2f(x);   // v_exp_f32: 2^x
#else
  return exp2f(x);                    // host-side parse only
#endif
}
// v_cos_f32 / v_sin_f32 take "revolutions": cosr(t) = cos(2*pi*t)
__device__ __forceinline__ float fcosr(float t) {
#if defined(__HIP_DEVICE_COMPILE__)
  return __builtin_amdgcn_cosf(t);
#else
  return cosf(t);
#endif
}
__device__ __forceinline__ float fsinr(float t) {
#if defined(__HIP_DEVICE_COMPILE__)
  return __builtin_amdgcn_sinf(t);
#else
  return sinf(t);
#endif
}

// ---------------------------------------------------------------------------
// Feature precompute: for each row of x (side 0) and y (side 1), for each q:
//   E    = exp(-2*pi^2 * sum_d p_d^2 * v_qd)
//   u_d  = p_d * mu_qd                      (revolution units)
//   For the 8 sign patterns s over d=1..3:
//     U = u0 + s1*u1 + s2*u2 + s3*u3
//     F[row][q*16 + 2s]   = scale * cos(2*pi*U)
//     F[row][q*16 + 2s+1] = scale * sin(2*pi*U)
//   scale = E * w_q/8 (x side) or E (y side)
// ---------------------------------------------------------------------------
__global__ __launch_bounds__(256) void sm_features(
    const float* __restrict__ x, const float* __restrict__ y,
    const float* __restrict__ lw, const float* __restrict__ mu,
    const float* __restrict__ lv,
    float* __restrict__ Fx, float* __restrict__ Fy)
{
  int gid = blockIdx.x * blockDim.x + threadIdx.x;   // 0 .. 8191
  int side = gid >> 12;
  int row  = gid & (NN - 1);
  const float* p = (side ? y : x) + row * 4;
  float p0 = p[0], p1 = p[1], p2 = p[2], p3 = p[3];
  float* F = (side ? Fy : Fx) + row * 64;

#pragma unroll
  for (int q = 0; q < 4; ++q) {
    float v0 = fexp2(kLOG2E * lv[q * 4 + 0]);
    float v1 = fexp2(kLOG2E * lv[q * 4 + 1]);
    float v2 = fexp2(kLOG2E * lv[q * 4 + 2]);
    float v3 = fexp2(kLOG2E * lv[q * 4 + 3]);
    float s  = p0 * p0 * v0 + p1 * p1 * v1 + p2 * p2 * v2 + p3 * p3 * v3;
    float E  = fexp2(kNEG_2PI2_L2E * s);
    float scale = E;
    if (side == 0) scale *= 0.125f * fexp2(kLOG2E * lw[q]);
    float u0 = p0 * mu[q * 4 + 0];
    float u1 = p1 * mu[q * 4 + 1];
    float u2 = p2 * mu[q * 4 + 2];
    float u3 = p3 * mu[q * 4 + 3];
#pragma unroll
    for (int sp = 0; sp < 8; ++sp) {
      float U = u0 + ((sp & 1) ? -u1 : u1)
                   + ((sp & 2) ? -u2 : u2)
                   + ((sp & 4) ? -u3 : u3);
      F[q * 16 + 2 * sp]     = scale * fcosr(U);
      F[q * 16 + 2 * sp + 1] = scale * fsinr(U);
    }
  }
}

// ---------------------------------------------------------------------------
// Main kernel: each wave computes a 32x32 output macro-tile (2x2 WMMA tiles).
// Per q:
//   G_ij = WMMA_f32_16x16x4( x_tile_i , (4pi^2*log2e * v_q) .* y_tile_j )
//   P_ij = sum over 4 K-chunks of WMMA( Fx chunk , Fy chunk )   (rank-16)
//   acc  += exp2(G) * P      (only transcendental left: 4 v_exp per element)
// ---------------------------------------------------------------------------
__global__ __launch_bounds__(256) void sm_main(
    const float* __restrict__ x, const float* __restrict__ y,
    const float* __restrict__ lv,
    const float* __restrict__ Fx, const float* __restrict__ Fy,
    float* __restrict__ out)
{
  const int lane = threadIdx.x & 31;
  const int wave = threadIdx.x >> 5;
  const int t  = blockIdx.x * 8 + wave;       // 0 .. 16383
  const int tn = t >> 7;                      // 128 macro-tiles per dim
  const int tm = t & 127;
  const int n0 = tn * 32;
  const int m0 = tm * 32;
  const int col  = lane & 15;
  const int half = lane >> 4;
  const int k0   = half * 2;                  // K slice held by this lane

  // A fragments (x rows) for the two row sub-tiles: {x[n][k0], x[n][k0+1]}
  v2f ax[2];
  ax[0] = *(const v2f*)(x + (n0 + col) * 4 + k0);
  ax[1] = *(const v2f*)(x + (n0 + 16 + col) * 4 + k0);
  // y rows for the two column sub-tiles
  v2f yb[2];
  yb[0] = *(const v2f*)(y + (m0 + col) * 4 + k0);
  yb[1] = *(const v2f*)(y + (m0 + 16 + col) * 4 + k0);

  const float* fx0 = Fx + (n0 + col) * 64 + k0;
  const float* fx1 = fx0 + 16 * 64;
  const float* fy0 = Fy + (m0 + col) * 64 + k0;
  const float* fy1 = fy0 + 16 * 64;

  v8f acc[2][2] = {};

#pragma unroll
  for (int q = 0; q < 4; ++q) {
    // B fragment for the Gaussian cross term, with 4pi^2*log2e folded in
    float va = fexp2(kLOG2E * lv[q * 4 + k0]);
    float vb = fexp2(kLOG2E * lv[q * 4 + k0 + 1]);
    float ca = kPOS_4PI2_L2E * va;
    float cb = kPOS_4PI2_L2E * vb;
    v2f bg[2];
    bg[0].x = ca * yb[0].x;  bg[0].y = cb * yb[0].y;
    bg[1].x = ca * yb[1].x;  bg[1].y = cb * yb[1].y;

    const v8f z = {};
    v8f G[2][2];
#pragma unroll
    for (int i = 0; i < 2; ++i)
#pragma unroll
      for (int j = 0; j < 2; ++j)
        G[i][j] = __builtin_amdgcn_wmma_f32_16x16x4_f32(
            false, ax[i], false, bg[j], (short)0, z, false, false);

    v8f P[2][2] = {};
#pragma unroll
    for (int c = 0; c < 4; ++c) {            // rank-16 = 4 chained K=4 WMMAs
      v2f af0 = *(const v2f*)(fx0 + q * 16 + c * 4);
      v2f af1 = *(const v2f*)(fx1 + q * 16 + c * 4);
      v2f bf0 = *(const v2f*)(fy0 + q * 16 + c * 4);
      v2f bf1 = *(const v2f*)(fy1 + q * 16 + c * 4);
      P[0][0] = __builtin_amdgcn_wmma_f32_16x16x4_f32(false, af0, false, bf0, (short)0, P[0][0], false, false);
      P[0][1] = __builtin_amdgcn_wmma_f32_16x16x4_f32(false, af0, false, bf1, (short)0, P[0][1], false, false);
      P[1][0] = __builtin_amdgcn_wmma_f32_16x16x4_f32(false, af1, false, bf0, (short)0, P[1][0], false, false);
      P[1][1] = __builtin_amdgcn_wmma_f32_16x16x4_f32(false, af1, false, bf1, (short)0, P[1][1], false, false);
    }

#pragma unroll
    for (int i = 0; i < 2; ++i)
#pragma unroll
      for (int j = 0; j < 2; ++j)
#pragma unroll
        for (int r = 0; r < 8; ++r)
          acc[i][j][r] += fexp2(G[i][j][r]) * P[i][j][r];
  }

  // D-matrix layout: VGPR r, lanes 0-15 -> row r, lanes 16-31 -> row r+8
#pragma unroll
  for (int i = 0; i < 2; ++i)
#pragma unroll
    for (int r = 0; r < 8; ++r) {
      int row = n0 + i * 16 + r + 8 * half;
      float* o = out + (size_t)row * MM + m0 + col;
      o[0]  = acc[i][0][r];
      o[16] = acc[i][1][r];
    }
}

extern "C" void kernel_launch(void* const* d_in, const int* in_sizes, int n_in,
                              void* d_out, int out_size, void* d_ws, size_t ws_size,
                              hipStream_t stream) {
  const float* x  = (const float*)d_in[0];   // [4096,4]
  const float* y  = (const float*)d_in[1];   // [4096,4]
  const float* lw = (const float*)d_in[2];   // [4]
  const float* mu = (const float*)d_in[3];   // [4,4]
  const float* lv = (const float*)d_in[4];   // [4,4]
  float* out = (float*)d_out;                // [4096,4096]

  float* Fx = (float*)d_ws;                  // [4096][64]  (1 MB)
  float* Fy = Fx + (size_t)NN * 64;          // [4096][64]  (1 MB)

  sm_features<<<(2 * NN) / 256, 256, 0, stream>>>(x, y, lw, mu, lv, Fx, Fy);
  sm_main<<<(128 * 128) / 8, 256, 0, stream>>>(x, y, lv, Fx, Fy, out);
}